// Attention_76141180223851
// MI455X (gfx1250) — compile-verified
//
#include <hip/hip_runtime.h>
#include <math.h>

typedef __attribute__((ext_vector_type(2))) float v2f;
typedef __attribute__((ext_vector_type(4))) float v4f;
typedef __attribute__((ext_vector_type(8))) float v8f;

#define SS 256
#define DD 64

// grid: B*H*(S/16) = 256 blocks, 256 threads (8 waves, wave32)
__global__ __launch_bounds__(256)
void Attention_76141180223851_kernel(const float* __restrict__ Q,
                                     const float* __restrict__ K,
                                     const float* __restrict__ V,
                                     const float* __restrict__ KS,
                                     const float* __restrict__ VS,
                                     const float* __restrict__ MASK,
                                     float* __restrict__ OUT,   // [B,H,S,D]
                                     float* __restrict__ ATT)   // [B,H,S,S]
{
    __shared__ float Qs[16][DD];        // 4 KB  query tile
    __shared__ float Sc[16][SS];        // 16 KB scores -> attn
    __shared__ float Op[2][16][DD];     // 8 KB  attn@V partials (two k-halves)

    const int t    = threadIdx.x;
    const int wave = t >> 5;
    const int lane = t & 31;
    const int bh   = blockIdx.x >> 4;    // 0..15  (b*H + h)
    const int qt   = blockIdx.x & 15;    // query tile index
    const int bidx = bh >> 3;            // batch (H = 8)

    const size_t qkbase = (size_t)bh * SS * DD;        // Q/K/V per-head base
    const size_t stbase = (size_t)bh * SS * SS * DD;   // KS/VS per-head base

    const int half16 = lane >> 4;        // 0 | 1  (low/high half-wave)
    const int l16    = lane & 15;
    const int koff   = half16 * 2;       // f32 WMMA K-lane offset (K=0,1 / K=2,3)

    // ---- stage Q tile (16 x 64) into LDS, one float4 per thread ----
    {
        const float4* src = (const float4*)(Q + qkbase + (size_t)qt * 16 * DD);
        ((float4*)&Qs[0][0])[t] = src[t];
    }
    __syncthreads();

    // ---- Phase 1a: scores = Q @ K^T via v_wmma_f32_16x16x4_f32 ----
    for (int kt2 = 0; kt2 < 2; ++kt2) {
        const int kt = wave * 2 + kt2;                       // k-tile 0..15
        v8f c = {0.f, 0.f, 0.f, 0.f, 0.f, 0.f, 0.f, 0.f};
        const float* Krow = K + qkbase + (size_t)(kt * 16 + l16) * DD;
#pragma unroll
        for (int s4 = 0; s4 < 16; ++s4) {
            const int dbase = s4 * 4 + koff;
            v2f a;  a.x  = Qs[l16][dbase];  a.y  = Qs[l16][dbase + 1];   // A: Q[m=l16][K]
            v2f bm; bm.x = Krow[dbase];     bm.y = Krow[dbase + 1];      // B: K^T[K][n=l16]
            c = __builtin_amdgcn_wmma_f32_16x16x4_f32(false, a, false, bm,
                                                      (short)0, c, false, false);
        }
#pragma unroll
        for (int r = 0; r < 8; ++r)
            Sc[r + half16 * 8][kt * 16 + l16] = c[r];
    }
    __syncthreads();

    // ---- Phase 1b: scores += einsum(q,d ; q,k,d) — stream KS once (non-temporal) ----
    // 4096 rows (16q x 256k), 2 rows per wave-iteration, 16 lanes x float4 per row
    for (int j = 0; j < 256; ++j) {
        const int row = wave * 512 + j * 2 + half16;
        const int ql  = row >> 8;          // local q 0..15
        const int kk  = row & 255;
        const int qg  = qt * 16 + ql;
        const v4f kv = __builtin_nontemporal_load(
            (const v4f*)(KS + stbase + ((size_t)qg * SS + kk) * DD + l16 * 4));
        const float4 qv = *(const float4*)&Qs[ql][l16 * 4];
        float p = kv.x * qv.x + kv.y * qv.y + kv.z * qv.z + kv.w * qv.w;
        p += __shfl_xor(p, 1, 32);
        p += __shfl_xor(p, 2, 32);
        p += __shfl_xor(p, 4, 32);
        p += __shfl_xor(p, 8, 32);
        if (l16 == 0) Sc[ql][kk] += p;
    }
    __syncthreads();

    // ---- Phase 1c: mask + scale + softmax over k; emit attn (NT store) ----
    for (int rr = 0; rr < 2; ++rr) {
        const int ql = wave * 2 + rr;
        const int qg = qt * 16 + ql;
        const float mval = (1.0f - MASK[bidx * SS + qg]) * -100000.0f;
        float z[8];
        float mx = -3.402823466e38f;
#pragma unroll
        for (int i = 0; i < 8; ++i) {
            z[i] = (Sc[ql][lane + i * 32] + mval) * 0.125f;   // 1/sqrt(64)
            mx = fmaxf(mx, z[i]);
        }
        mx = fmaxf(mx, __shfl_xor(mx, 1, 32));
        mx = fmaxf(mx, __shfl_xor(mx, 2, 32));
        mx = fmaxf(mx, __shfl_xor(mx, 4, 32));
        mx = fmaxf(mx, __shfl_xor(mx, 8, 32));
        mx = fmaxf(mx, __shfl_xor(mx, 16, 32));
        float sum = 0.f;
#pragma unroll
        for (int i = 0; i < 8; ++i) { z[i] = __expf(z[i] - mx); sum += z[i]; }
        sum += __shfl_xor(sum, 1, 32);
        sum += __shfl_xor(sum, 2, 32);
        sum += __shfl_xor(sum, 4, 32);
        sum += __shfl_xor(sum, 8, 32);
        sum += __shfl_xor(sum, 16, 32);
        const float inv = 1.0f / sum;
        float* arow = ATT + ((size_t)bh * SS + qg) * SS;
#pragma unroll
        for (int i = 0; i < 8; ++i) {
            const float av = z[i] * inv;
            Sc[ql][lane + i * 32] = av;
            __builtin_nontemporal_store(av, arow + lane + i * 32);
        }
    }
    __syncthreads();

    // ---- Phase 2: attn @ V via WMMA (waves split 4 d-tiles x 2 k-halves) ----
    {
        const int dt    = wave & 3;
        const int halfk = wave >> 2;
        const int kb0   = halfk * 128;
        v8f c = {0.f, 0.f, 0.f, 0.f, 0.f, 0.f, 0.f, 0.f};
#pragma unroll
        for (int s4 = 0; s4 < 32; ++s4) {
            const int kbase = kb0 + s4 * 4 + koff;
            v2f a;  a.x = Sc[l16][kbase]; a.y = Sc[l16][kbase + 1];      // A: attn[m][K]
            const float* vp = V + qkbase + (size_t)kbase * DD + dt * 16 + l16;
            v2f bm; bm.x = vp[0]; bm.y = vp[DD];                         // B: V[K][n]
            c = __builtin_amdgcn_wmma_f32_16x16x4_f32(false, a, false, bm,
                                                      (short)0, c, false, false);
        }
#pragma unroll
        for (int r = 0; r < 8; ++r)
            Op[halfk][r + half16 * 8][dt * 16 + l16] = c[r];
    }
    __syncthreads();

    // ---- Phase 3: out = attn@V + sum_k attn[q,k]*VS[q,k,:] — stream VS once (NT) ----
    {
        const int ql    = wave * 2 + half16;   // waves cover q 0..15
        const int qg    = qt * 16 + ql;
        const int dbase = l16 * 4;
        const float* vsp = VS + stbase + (size_t)qg * SS * DD + dbase;
        float ax = 0.f, ay = 0.f, az = 0.f, aw = 0.f;
#pragma unroll 8
        for (int k = 0; k < SS; ++k) {
            const float s = Sc[ql][k];
            const v4f  v = __builtin_nontemporal_load((const v4f*)(vsp + (size_t)k * DD));
            ax = fmaf(s, v.x, ax);
            ay = fmaf(s, v.y, ay);
            az = fmaf(s, v.z, az);
            aw = fmaf(s, v.w, aw);
        }
        v4f o;
        o.x = ax + Op[0][ql][dbase + 0] + Op[1][ql][dbase + 0];
        o.y = ay + Op[0][ql][dbase + 1] + Op[1][ql][dbase + 1];
        o.z = az + Op[0][ql][dbase + 2] + Op[1][ql][dbase + 2];
        o.w = aw + Op[0][ql][dbase + 3] + Op[1][ql][dbase + 3];
        __builtin_nontemporal_store(o, (v4f*)(OUT + ((size_t)bh * SS + qg) * DD + dbase));
    }
}

extern "C" void kernel_launch(void* const* d_in, const int* in_sizes, int n_in,
                              void* d_out, int out_size, void* d_ws, size_t ws_size,
                              hipStream_t stream) {
    const float* Q    = (const float*)d_in[0];
    const float* K    = (const float*)d_in[1];
    const float* V    = (const float*)d_in[2];
    const float* KS   = (const float*)d_in[3];
    const float* VS   = (const float*)d_in[4];
    const float* MASK = (const float*)d_in[5];

    float* OUT = (float*)d_out;                  // [2,8,256,64]  = 262144 floats
    float* ATT = OUT + 2 * 8 * 256 * 64;         // [2,8,256,256] follows in tuple order

    // B*H*(S/16) = 2*8*16 = 256 workgroups, 256 threads each (8 wave32)
    Attention_76141180223851_kernel<<<dim3(256), dim3(256), 0, stream>>>(
        Q, K, V, KS, VS, MASK, OUT, ATT);
}